// GraphModel_11836929868640
// MI455X (gfx1250) — compile-verified
//
#include <hip/hip_runtime.h>
#include <math.h>

typedef __attribute__((ext_vector_type(2))) float v2f;
typedef __attribute__((ext_vector_type(8))) float v8f;

#define HDIM 32
#define NEG_SLOPE 0.2f

// ---- ordered-uint encoding for float atomic max (handles negatives) ----
__device__ __forceinline__ unsigned f2ord(float f) {
    unsigned u = __float_as_uint(f);
    return (u & 0x80000000u) ? ~u : (u | 0x80000000u);
}
__device__ __forceinline__ float ord2f(unsigned k) {
    unsigned u = (k & 0x80000000u) ? (k & 0x7FFFFFFFu) : ~k;
    return __uint_as_float(u);
}

// ============================================================
// h = X[N,Din] @ W[Din,32] via V_WMMA_F32_16X16X4_F32
// One wave per 16x16 output tile. grid = (Npad/16, 2)
// A 16x4 layout: lane L holds row L&15; lanes<16 -> K=k0,k0+1 ; lanes>=16 -> K=k0+2,k0+3
// B 4x16 layout: lane L holds col L&15; vgpr0=K{0|2}, vgpr1=K{1|3}
// C/D: vgpr i -> row i + (L>>4)*8, col L&15
// ============================================================
__global__ void gemm_wmma_kernel(const float* __restrict__ X,
                                 const float* __restrict__ W,
                                 float* __restrict__ Hout,
                                 int N, int Din) {
    const int lane = threadIdx.x;          // 0..31
    const int row0 = blockIdx.x * 16;
    const int col0 = blockIdx.y * 16;
    const int r    = lane & 15;
    const int kh   = (lane >> 4) * 2;

    int arow = row0 + r; if (arow >= N) arow = N - 1;   // clamp tail reads
    const int col = col0 + r;

    v8f acc = {};
    for (int k0 = 0; k0 < Din; k0 += 4) {
        v2f a, b;
        a.x = X[(size_t)arow * Din + k0 + kh];
        a.y = X[(size_t)arow * Din + k0 + kh + 1];
        b.x = W[(size_t)(k0 + kh)     * HDIM + col];
        b.y = W[(size_t)(k0 + kh + 1) * HDIM + col];
        acc = __builtin_amdgcn_wmma_f32_16x16x4_f32(
            /*neg_a=*/false, a, /*neg_b=*/false, b,
            /*c_mod=*/(short)0, acc, /*reuse_a=*/false, /*reuse_b=*/false);
    }
    const int rbase = row0 + (lane >> 4) * 8;
    #pragma unroll
    for (int i = 0; i < 8; ++i) {
        Hout[(size_t)(rbase + i) * HDIM + col0 + (lane & 15)] = acc[i];
    }
}

// alpha_s/alpha_d per node; init m (ordered -inf), z=0, zero output accumulator
__global__ void alpha_init_kernel(const float* __restrict__ h,
                                  const float* __restrict__ a_s,
                                  const float* __restrict__ a_d,
                                  float* __restrict__ as_, float* __restrict__ ad_,
                                  unsigned* __restrict__ mkey, float* __restrict__ z,
                                  float* __restrict__ hout, int N) {
    int n = blockIdx.x * blockDim.x + threadIdx.x;
    if (n >= N) return;
    float s0 = 0.f, s1 = 0.f;
    #pragma unroll
    for (int j = 0; j < HDIM; ++j) {
        float v = h[(size_t)n * HDIM + j];
        s0 += v * a_s[j];
        s1 += v * a_d[j];
        hout[(size_t)n * HDIM + j] = 0.f;
    }
    as_[n] = s0; ad_[n] = s1;
    mkey[n] = f2ord(-3.0e38f);
    z[n] = 0.f;
}

// pass A: e = leaky_relu(as[src]+ad[dst]); segment max into mkey[dst]
__global__ void edgeA_kernel(const int* __restrict__ src, const int* __restrict__ dst,
                             const float* __restrict__ as_, const float* __restrict__ ad_,
                             float* __restrict__ e, unsigned* __restrict__ mkey,
                             int E, int N) {
    int t = blockIdx.x * blockDim.x + threadIdx.x;
    int tot = E + N;
    if (t >= tot) return;
    int s, d;
    if (t < E) { s = src[t]; d = dst[t]; } else { s = d = t - E; }
    float v = as_[s] + ad_[d];
    v = v > 0.f ? v : NEG_SLOPE * v;
    e[t] = v;
    atomicMax(&mkey[d], f2ord(v));
}

// pass B: w = exp(e - m[dst]); z[dst] += w; store w into e
__global__ void edgeB_kernel(const int* __restrict__ dst,
                             float* __restrict__ e,
                             const unsigned* __restrict__ mkey,
                             float* __restrict__ z, int E, int N) {
    int t = blockIdx.x * blockDim.x + threadIdx.x;
    int tot = E + N;
    if (t >= tot) return;
    int d = (t < E) ? dst[t] : (t - E);
    float w = expf(e[t] - ord2f(mkey[d]));
    e[t] = w;
    atomicAdd(&z[d], w);
}

// pass C: hout[dst] += h[src] * (w / (z[dst]+1e-16)); one lane per (edge, feature)
__global__ void edgeC_kernel(const int* __restrict__ src, const int* __restrict__ dst,
                             const float* __restrict__ e, const float* __restrict__ z,
                             const float* __restrict__ h, float* __restrict__ hout,
                             int E, int N) {
    long long t = (long long)blockIdx.x * blockDim.x + threadIdx.x;
    long long tot = (long long)(E + N) * HDIM;
    if (t >= tot) return;
    int edge = (int)(t >> 5);
    int j    = (int)(t & 31);
    int s, d;
    if (edge < E) { s = src[edge]; d = dst[edge]; } else { s = d = edge - E; }
    float alpha = e[edge] / (z[d] + 1e-16f);
    atomicAdd(&hout[(size_t)d * HDIM + j], h[(size_t)s * HDIM + j] * alpha);
}

// out += bias; optional exact GELU
__global__ void biasact_kernel(float* __restrict__ hout, const float* __restrict__ b,
                               int N, int doGelu) {
    int t = blockIdx.x * blockDim.x + threadIdx.x;
    if (t >= N * HDIM) return;
    int j = t & 31;
    float v = hout[t] + b[j];
    if (doGelu) v = 0.5f * v * (1.0f + erff(v * 0.70710678118654752f));
    hout[t] = v;
}

// ---------------- pooling ----------------
__global__ void pool_init_kernel(float* __restrict__ gsum, unsigned* __restrict__ gmaxk,
                                 float* __restrict__ cnt, int B) {
    int t = blockIdx.x * blockDim.x + threadIdx.x;
    if (t >= B * HDIM) return;
    gsum[t]  = 0.f;
    gmaxk[t] = f2ord(-3.0e38f);
    if (t < B) cnt[t] = 0.f;
}

__global__ void pool_acc_kernel(const float* __restrict__ h, const int* __restrict__ batch,
                                float* __restrict__ gsum, unsigned* __restrict__ gmaxk,
                                float* __restrict__ cnt, int N) {
    int t = blockIdx.x * blockDim.x + threadIdx.x;
    if (t >= N * HDIM) return;
    int n = t >> 5, j = t & 31;
    int b = batch[n];
    float v = h[t];
    atomicAdd(&gsum[b * HDIM + j], v);
    atomicMax(&gmaxk[b * HDIM + j], f2ord(v));
    if (j == 0) atomicAdd(&cnt[b], 1.f);
}

__global__ void gvec_kernel(const float* __restrict__ gsum, const unsigned* __restrict__ gmaxk,
                            const float* __restrict__ cnt, float* __restrict__ gvec, int B) {
    int t = blockIdx.x * blockDim.x + threadIdx.x;
    if (t >= B * 64) return;
    int b = t >> 6, j = t & 63;
    gvec[t] = (j < 32) ? gsum[b * 32 + j] / fmaxf(cnt[b], 1.f)
                       : ord2f(gmaxk[b * 32 + (j - 32)]);
}

// ---------------- MLP head (B=1024 rows; negligible cost) ----------------
__global__ void head_kernel(const float* __restrict__ gvec,
                            const float* __restrict__ Wr, const float* __restrict__ br,
                            const float* __restrict__ Wm0, const float* __restrict__ bm0,
                            const float* __restrict__ Wm1, const float* __restrict__ bm1,
                            const float* __restrict__ Wl, const float* __restrict__ bl,
                            float* __restrict__ out, int B) {
    int b = blockIdx.x * blockDim.x + threadIdx.x;
    if (b >= B) return;
    const float* g = gvec + (size_t)b * 64;
    float t1[32], t2[32], t3[32];
    for (int j = 0; j < 32; ++j) {
        float s = br[j];
        for (int k = 0; k < 64; ++k) s += g[k] * Wr[k * 32 + j];
        t1[j] = s;
    }
    for (int j = 0; j < 32; ++j) {
        float s = bm0[j];
        for (int k = 0; k < 32; ++k) s += t1[k] * Wm0[k * 32 + j];
        t2[j] = fmaxf(s, 0.f);
    }
    for (int j = 0; j < 32; ++j) {
        float s = bm1[j];
        for (int k = 0; k < 32; ++k) s += t2[k] * Wm1[k * 32 + j];
        t3[j] = fmaxf(s, 0.f);
    }
    for (int j = 0; j < 48; ++j) {
        float s = bl[j];
        for (int k = 0; k < 32; ++k) s += t3[k] * Wl[k * 48 + j];
        out[(size_t)b * 48 + j] = s;
    }
}

extern "C" void kernel_launch(void* const* d_in, const int* in_sizes, int n_in,
                              void* d_out, int out_size, void* d_ws, size_t ws_size,
                              hipStream_t stream) {
    const float* x     = (const float*)d_in[0];
    const int*   ei    = (const int*)  d_in[1];
    const int*   batch = (const int*)  d_in[2];
    // d_in[3] = num_graphs (device scalar, shapes are static) -> B=1024
    const float* W1  = (const float*)d_in[4];
    const float* as1 = (const float*)d_in[5];
    const float* ad1 = (const float*)d_in[6];
    const float* b1  = (const float*)d_in[7];
    const float* W2  = (const float*)d_in[8];
    const float* as2 = (const float*)d_in[9];
    const float* ad2 = (const float*)d_in[10];
    const float* b2  = (const float*)d_in[11];
    const float* W3  = (const float*)d_in[12];
    const float* as3 = (const float*)d_in[13];
    const float* ad3 = (const float*)d_in[14];
    const float* b3  = (const float*)d_in[15];
    const float* Wr  = (const float*)d_in[16];
    const float* br  = (const float*)d_in[17];
    const float* Wm0 = (const float*)d_in[18];
    const float* bm0 = (const float*)d_in[19];
    const float* Wm1 = (const float*)d_in[20];
    const float* bm1 = (const float*)d_in[21];
    const float* Wl  = (const float*)d_in[22];
    const float* bl  = (const float*)d_in[23];

    const int N = in_sizes[0] / 16;     // 50000
    const int E = in_sizes[1] / 2;      // 1600000
    const int B = 1024;
    const int Npad = (N + 15) & ~15;

    // workspace carve-up (floats)
    float* ws = (float*)d_ws;
    float*    h     = ws;                 ws += (size_t)Npad * HDIM;
    float*    hagg  = ws;                 ws += (size_t)Npad * HDIM;
    float*    as_   = ws;                 ws += N;
    float*    ad_   = ws;                 ws += N;
    unsigned* mkey  = (unsigned*)ws;      ws += N;
    float*    z     = ws;                 ws += N;
    float*    e     = ws;                 ws += (size_t)(E + N);
    float*    gsum  = ws;                 ws += B * HDIM;
    unsigned* gmaxk = (unsigned*)ws;      ws += B * HDIM;
    float*    cnt   = ws;                 ws += B;
    float*    gvec  = ws;                 ws += B * 64;

    const int* src = ei;
    const int* dst = ei + E;

    const int tot  = E + N;
    const int TB   = 256;

    auto run_layer = [&](const float* xin, int Din, const float* W,
                         const float* a_s, const float* a_d, const float* bb, int doGelu) {
        dim3 g1(Npad / 16, 2);
        gemm_wmma_kernel<<<g1, 32, 0, stream>>>(xin, W, h, N, Din);
        alpha_init_kernel<<<(N + TB - 1) / TB, TB, 0, stream>>>(h, a_s, a_d, as_, ad_, mkey, z, hagg, N);
        edgeA_kernel<<<(tot + TB - 1) / TB, TB, 0, stream>>>(src, dst, as_, ad_, e, mkey, E, N);
        edgeB_kernel<<<(tot + TB - 1) / TB, TB, 0, stream>>>(dst, e, mkey, z, E, N);
        long long tc = (long long)tot * HDIM;
        edgeC_kernel<<<(unsigned)((tc + TB - 1) / TB), TB, 0, stream>>>(src, dst, e, z, h, hagg, E, N);
        biasact_kernel<<<(N * HDIM + TB - 1) / TB, TB, 0, stream>>>(hagg, bb, N, doGelu);
    };

    run_layer(x,    16, W1, as1, ad1, b1, 1);
    run_layer(hagg, 32, W2, as2, ad2, b2, 1);
    run_layer(hagg, 32, W3, as3, ad3, b3, 0);

    pool_init_kernel<<<(B * HDIM + TB - 1) / TB, TB, 0, stream>>>(gsum, gmaxk, cnt, B);
    pool_acc_kernel<<<(N * HDIM + TB - 1) / TB, TB, 0, stream>>>(hagg, batch, gsum, gmaxk, cnt, N);
    gvec_kernel<<<(B * 64 + TB - 1) / TB, TB, 0, stream>>>(gsum, gmaxk, cnt, gvec, B);
    head_kernel<<<(B + 63) / 64, 64, 0, stream>>>(gvec, Wr, br, Wm0, bm0, Wm1, bm1, Wl, bl,
                                                  (float*)d_out, B);
}